// GaussianAttention_6768868458579
// MI455X (gfx1250) — compile-verified
//
#include <hip/hip_runtime.h>
#include <hip/hip_bf16.h>

typedef __attribute__((ext_vector_type(2))) float v2f;
typedef __attribute__((ext_vector_type(8))) float v8f;

#define BATCH 64
#define TSTEPS 1024
#define HDIM 512
#define KMIX 10
#define UDIM 64
#define VDIM 80
#define NW 30          // 3*KMIX
#define NPAD 32        // padded abk width
#define OUTW 595       // H + V + S
#define MROWS (BATCH * TSTEPS)   // 65536

// ---------------------------------------------------------------------------
// Kernel 1: abk = exp(X @ W + b), fused with out[:,0:512] = X.
// One workgroup = 8 waves; each wave = one 16-row M tile; N=32 (2 WMMA tiles).
// fp32 WMMA 16x16x4: A layout lane l -> M=l%16, K = k + 2*(l/16) + {0,1};
// B layout lane l -> N=l%16, same K split; C/D: VGPR r -> M=r+8*(l/16), N=l%16.
// ---------------------------------------------------------------------------
__global__ __launch_bounds__(256) void abk_gemm_kernel(
    const float* __restrict__ x,     // [MROWS, HDIM]
    const float* __restrict__ w,     // [HDIM, NW]
    const float* __restrict__ bias,  // [NW]
    float* __restrict__ out,         // [MROWS, OUTW]
    float* __restrict__ abk)         // [MROWS, NPAD]
{
    __shared__ float wlds[HDIM * NPAD];   // 64 KB, zero-padded to 32 cols
    const int tid = threadIdx.x;
    for (int idx = tid; idx < HDIM * NPAD; idx += 256) {
        const int kk = idx >> 5, n = idx & 31;
        wlds[idx] = (n < NW) ? w[kk * NW + n] : 0.0f;
    }
    __syncthreads();

    const int lane  = tid & 31;
    const int wave  = tid >> 5;
    const int mbase = blockIdx.x * 128 + wave * 16;
    const int mrow  = mbase + (lane & 15);
    const int koff  = (lane >> 4) * 2;     // 0 for lanes 0-15, 2 for lanes 16-31
    const int nlow  = lane & 15;

    const float* __restrict__ xrow = x + (size_t)mrow * HDIM;
    float* __restrict__ orow = out + (size_t)mrow * OUTW;

    v8f acc0 = {};
    v8f acc1 = {};

    for (int k = 0; k < HDIM; k += 4) {
        const int kk = k + koff;
        const v2f a = *(const v2f*)(xrow + kk);       // 8B-aligned (kk even, HDIM even)
        // fused streaming copy x -> out cols [0,512)
        orow[kk]     = a.x;
        orow[kk + 1] = a.y;
        v2f b0, b1;
        b0.x = wlds[kk * NPAD + nlow];
        b0.y = wlds[(kk + 1) * NPAD + nlow];
        b1.x = wlds[kk * NPAD + 16 + nlow];
        b1.y = wlds[(kk + 1) * NPAD + 16 + nlow];
        acc0 = __builtin_amdgcn_wmma_f32_16x16x4_f32(false, a, false, b0,
                                                     (short)0, acc0, false, false);
        acc1 = __builtin_amdgcn_wmma_f32_16x16x4_f32(false, a, false, b1,
                                                     (short)0, acc1, false, false);
    }

    const int n1 = nlow + 16;
    const float bs0 = bias[nlow];                    // nlow < 16 <= 30, always valid
    const float bs1 = (n1 < NW) ? bias[n1] : 0.0f;
    const int rbase = mbase + (lane >> 4) * 8;
#pragma unroll
    for (int r = 0; r < 8; ++r) {
        const size_t m = (size_t)(rbase + r);
        abk[m * NPAD + nlow] = __expf(acc0[r] + bs0);
        abk[m * NPAD + n1]   = __expf(acc1[r] + bs1);  // padded cols -> exp(0)=1, never read
    }
}

// ---------------------------------------------------------------------------
// Kernel 2: in-place cumulative sum of kappa_hat over T (the scan carry).
// 640 independent sequences (b,k), one per thread.
// ---------------------------------------------------------------------------
__global__ void kappa_scan_kernel(const float* __restrict__ kappa0,  // [B, K]
                                  float* __restrict__ abk)           // [MROWS, NPAD]
{
    const int idx = blockIdx.x * blockDim.x + threadIdx.x;
    if (idx >= BATCH * KMIX) return;
    const int b = idx / KMIX, k = idx % KMIX;
    float acc = kappa0[idx];
    float* base = abk + (size_t)b * TSTEPS * NPAD + 20 + k;
    for (int t = 0; t < TSTEPS; ++t) {
        acc += base[(size_t)t * NPAD];
        base[(size_t)t * NPAD] = acc;
    }
}

// ---------------------------------------------------------------------------
// Kernel 3: phi on the fly + window = phi @ char[b] (fp32 WMMA), plus stroke
// copy.  One workgroup = (b, 128-t block); wave = 16-t tile; 5 N-tiles of 16.
// ---------------------------------------------------------------------------
__global__ __launch_bounds__(256) void window_kernel(
    const float* __restrict__ abk,    // [MROWS, NPAD]: alpha 0..9, beta 10..19, kappa 20..29
    const float* __restrict__ chr,    // [B, UDIM, VDIM]
    const float* __restrict__ stroke, // [MROWS, 3]
    float* __restrict__ out)          // [MROWS, OUTW]
{
    __shared__ float clds[UDIM * VDIM];   // 20 KB
    const int b    = blockIdx.x >> 3;
    const int tblk = blockIdx.x & 7;
    const int tid  = threadIdx.x;
    for (int idx = tid; idx < UDIM * VDIM; idx += 256)
        clds[idx] = chr[(size_t)b * UDIM * VDIM + idx];
    __syncthreads();

    const int lane  = tid & 31;
    const int wave  = tid >> 5;
    const int tbase = tblk * 128 + wave * 16;
    const int trow  = tbase + (lane & 15);
    const size_t m  = (size_t)b * TSTEPS + trow;
    const int nlow  = lane & 15;
    const int uoff  = (lane >> 4) * 2;

    float al[KMIX], be[KMIX], kp[KMIX];
    const float* __restrict__ row = abk + m * NPAD;
#pragma unroll
    for (int j = 0; j < KMIX; ++j) {
        al[j] = row[j];
        be[j] = row[10 + j];
        kp[j] = row[20 + j];
    }

    v8f acc[5];
    const v8f vzero = {};
#pragma unroll
    for (int nt = 0; nt < 5; ++nt) acc[nt] = vzero;

    for (int ks = 0; ks < 16; ++ks) {
        const int u0 = ks * 4 + uoff;
        const float fu0 = (float)u0, fu1 = (float)(u0 + 1);
        float p0 = 0.0f, p1 = 0.0f;
#pragma unroll
        for (int j = 0; j < KMIX; ++j) {
            const float d0 = kp[j] - fu0;
            const float d1 = kp[j] - fu1;
            p0 += al[j] * __expf(-be[j] * d0 * d0);
            p1 += al[j] * __expf(-be[j] * d1 * d1);
        }
        v2f a;
        a.x = p0;
        a.y = p1;
#pragma unroll
        for (int nt = 0; nt < 5; ++nt) {
            v2f bv;
            bv.x = clds[u0 * VDIM + nt * 16 + nlow];
            bv.y = clds[(u0 + 1) * VDIM + nt * 16 + nlow];
            acc[nt] = __builtin_amdgcn_wmma_f32_16x16x4_f32(false, a, false, bv,
                                                            (short)0, acc[nt], false, false);
        }
    }

    // window -> out cols [512, 592)
    const int rb = tbase + (lane >> 4) * 8;
#pragma unroll
    for (int nt = 0; nt < 5; ++nt) {
#pragma unroll
        for (int r = 0; r < 8; ++r) {
            const size_t mm = (size_t)b * TSTEPS + rb + r;
            out[mm * OUTW + HDIM + nt * 16 + nlow] = acc[nt][r];
        }
    }

    // stroke -> out cols [592, 595)
    if (lane < 16) {
        const size_t mm = (size_t)b * TSTEPS + tbase + lane;
        out[mm * OUTW + 592] = stroke[mm * 3 + 0];
        out[mm * OUTW + 593] = stroke[mm * 3 + 1];
        out[mm * OUTW + 594] = stroke[mm * 3 + 2];
    }
}

// ---------------------------------------------------------------------------
extern "C" void kernel_launch(void* const* d_in, const int* in_sizes, int n_in,
                              void* d_out, int out_size, void* d_ws, size_t ws_size,
                              hipStream_t stream) {
    (void)in_sizes; (void)n_in; (void)out_size; (void)ws_size;
    const float* x      = (const float*)d_in[0];  // [B,T,H]
    const float* stroke = (const float*)d_in[1];  // [B,T,S]
    const float* chr    = (const float*)d_in[2];  // [B,U,V]
    const float* w      = (const float*)d_in[3];  // [H,3K]
    const float* bias   = (const float*)d_in[4];  // [3K]
    const float* kappa0 = (const float*)d_in[5];  // [B,K,1]
    float* out = (float*)d_out;                   // [B,T,H+V+S]
    float* abk = (float*)d_ws;                    // [MROWS, NPAD] = 8 MB

    abk_gemm_kernel<<<MROWS / 128, 256, 0, stream>>>(x, w, bias, out, abk);
    kappa_scan_kernel<<<(BATCH * KMIX + 255) / 256, 256, 0, stream>>>(kappa0, abk);
    window_kernel<<<BATCH * 8, 256, 0, stream>>>(abk, chr, stroke, out);
}